// TrajDecoder_62577673503549
// MI455X (gfx1250) — compile-verified
//
#include <hip/hip_runtime.h>
#include <math.h>

// Problem constants (match reference)
#define Bn 1024
#define Sn 200
#define Hn 256
#define En 128
#define Vn 100000

typedef float v2f __attribute__((ext_vector_type(2)));
typedef float v8f __attribute__((ext_vector_type(8)));

// GCC-style vector type matching the async-LDS builtin's parameter type
typedef int v4i_b128 __attribute__((vector_size(16)));
typedef __attribute__((address_space(1))) v4i_b128 gbl_v4i;   // global (AS1)
typedef __attribute__((address_space(3))) v4i_b128 lds_v4i;   // LDS    (AS3)

#define LDSPAD 260   // 256 + 4-float pad: row stride 1040B (16B aligned, bank-skewed)

// ---------------------------------------------------------------------------
// Async global->LDS copy of 16 bytes (gfx1250 GLOBAL_LOAD_ASYNC_TO_LDS_B128).
// Falls back to a synchronous copy if the builtin is unavailable.
// ---------------------------------------------------------------------------
__device__ __forceinline__ void async_cp16(float* lds_dst, const float* gsrc) {
#if __has_builtin(__builtin_amdgcn_global_load_async_to_lds_b128)
    __builtin_amdgcn_global_load_async_to_lds_b128(
        (gbl_v4i*)gsrc, (lds_v4i*)lds_dst, 0, 0);
#else
    *(float4*)lds_dst = *(const float4*)gsrc;
#endif
}

__device__ __forceinline__ void async_wait_all() {
#if __has_builtin(__builtin_amdgcn_global_load_async_to_lds_b128)
#  if __has_builtin(__builtin_amdgcn_s_wait_asynccnt)
    __builtin_amdgcn_s_wait_asynccnt(0);
#  else
    asm volatile("s_wait_asynccnt 0" ::: "memory");
#  endif
#endif
}

// ---------------------------------------------------------------------------
// Generic fp32 WMMA GEMM: C[M,N] = A[M,K] * B + bias[n]
//   BT=false : Bm is (K,N) row-major;  BT=true : Bm is (N,K) row-major (A*B^T)
// Each wave computes a 16x32 tile. Used for the small GEMMs (hWa, gi, gh).
// ---------------------------------------------------------------------------
template <bool BT, bool NTSTORE>
__global__ void wmma_gemm_f32(const float* __restrict__ A,
                              const float* __restrict__ Bm,
                              const float* __restrict__ bias,
                              float* __restrict__ C,
                              int M, int N, int K,
                              int lda, int ldb, long ldc, int total_waves) {
    int wave = blockIdx.x * (blockDim.x >> 5) + (threadIdx.x >> 5);
    if (wave >= total_waves) return;                 // wave-uniform: EXEC stays all-ones
    int tiles_n = N >> 5;
    int m0 = (wave / tiles_n) << 4;
    int n0 = (wave % tiles_n) << 5;

    int lane = threadIdx.x & 31;
    int l  = lane & 15;
    int hl = lane >> 4;

    v8f c0 = {}; v8f c1 = {};
    const float* arow = A + (long)(m0 + l) * lda;

    for (int k0 = 0; k0 < K; k0 += 4) {
        v2f a;
        a.x = arow[k0 + 2 * hl + 0];
        a.y = arow[k0 + 2 * hl + 1];
        v2f b0, b1;
        if (BT) {
            const float* br0 = Bm + (long)(n0 + l) * ldb + k0 + 2 * hl;
            const float* br1 = Bm + (long)(n0 + 16 + l) * ldb + k0 + 2 * hl;
            b0.x = br0[0]; b0.y = br0[1];
            b1.x = br1[0]; b1.y = br1[1];
        } else {
            const float* br = Bm + (long)(k0 + 2 * hl) * ldb;
            b0.x = br[n0 + l];       b0.y = br[ldb + n0 + l];
            b1.x = br[n0 + 16 + l];  b1.y = br[ldb + n0 + 16 + l];
        }
        c0 = __builtin_amdgcn_wmma_f32_16x16x4_f32(false, a, false, b0, (short)0, c0, false, false);
        c1 = __builtin_amdgcn_wmma_f32_16x16x4_f32(false, a, false, b1, (short)0, c1, false, false);
    }

    float bias0 = bias[n0 + l];
    float bias1 = bias[n0 + 16 + l];
    #pragma unroll
    for (int j = 0; j < 8; ++j) {
        long row = m0 + j + 8 * hl;
        float* crow = C + row * ldc;
        float o0 = c0[j] + bias0;
        float o1 = c1[j] + bias1;
        if (NTSTORE) {
            __builtin_nontemporal_store(o0, &crow[n0 + l]);
            __builtin_nontemporal_store(o1, &crow[n0 + 16 + l]);
        } else {
            crow[n0 + l] = o0;
            crow[n0 + 16 + l] = o1;
        }
    }
}

// ---------------------------------------------------------------------------
// Vocab projection: pred[B,V] = h_new @ W_out^T + b_out  (M=1024,N=100000,K=256)
// Block = 256 threads = 8 waves -> 128(M) x 32(N) macro-tile.
// The 32x256 W_out tile (33 KB) is staged into LDS once per block with
// async b128 loads (ASYNCcnt), then the K-loop reads B from LDS (bank-skewed
// via row padding) while A (L2-resident h_new) streams from global.
// Output uses nontemporal stores so the 410 MB pred stream doesn't evict L2.
// ---------------------------------------------------------------------------
__global__ void pred_gemm_kernel(const float* __restrict__ A,     // (B,H) h_new
                                 const float* __restrict__ Bt,    // (V,H) W_out
                                 const float* __restrict__ bias,  // (V,)
                                 float* __restrict__ C) {         // (B,V)
    __shared__ float lsB[32 * LDSPAD];                 // 33.3 KB

    const int tiles_n = Vn / 32;                       // 3125
    int mblk = blockIdx.x / tiles_n;                   // 0..7
    int n0   = (blockIdx.x % tiles_n) * 32;

    // stage B tile: 32 rows x 64 chunks(16B) = 2048 chunks, 8 per thread
    for (int c = threadIdx.x; c < 32 * 64; c += 256) {
        int n  = c >> 6;
        int kc = (c & 63) << 2;
        async_cp16(&lsB[n * LDSPAD + kc], Bt + (long)(n0 + n) * Hn + kc);
    }
    async_wait_all();
    __syncthreads();

    int wv   = threadIdx.x >> 5;
    int m0   = mblk * 128 + wv * 16;
    int lane = threadIdx.x & 31;
    int l    = lane & 15;
    int hl   = lane >> 4;

    v8f c0 = {}; v8f c1 = {};
    const float* arow = A + (long)(m0 + l) * Hn;
    const float* br0  = &lsB[l * LDSPAD];
    const float* br1  = &lsB[(16 + l) * LDSPAD];

    #pragma unroll 8
    for (int k0 = 0; k0 < Hn; k0 += 4) {
        int kk = k0 + 2 * hl;
        v2f a;  a.x  = arow[kk]; a.y  = arow[kk + 1];
        v2f b0; b0.x = br0[kk];  b0.y = br0[kk + 1];
        v2f b1; b1.x = br1[kk];  b1.y = br1[kk + 1];
        c0 = __builtin_amdgcn_wmma_f32_16x16x4_f32(false, a, false, b0, (short)0, c0, false, false);
        c1 = __builtin_amdgcn_wmma_f32_16x16x4_f32(false, a, false, b1, (short)0, c1, false, false);
    }

    float bias0 = bias[n0 + l];
    float bias1 = bias[n0 + 16 + l];
    #pragma unroll
    for (int j = 0; j < 8; ++j) {
        long row = m0 + j + 8 * hl;
        float* crow = C + row * (long)Vn;
        __builtin_nontemporal_store(c0[j] + bias0, &crow[n0 + l]);
        __builtin_nontemporal_store(c1[j] + bias1, &crow[n0 + 16 + l]);
    }
}

// ---------------------------------------------------------------------------
// Fused attention score: for 16 rows of enc (flattened (s*B+b, 256)):
//   energy = tanh(enc_row @ Wa_bot + hWa[b])   (hWa already contains ba)
//   att[b,s] = sum_n energy[n] * v[n]          (masked)
// Block = 512 threads = 16 waves; wave w handles N-tile n0=w*16 (N=256).
// The shared 16x256 enc tile is staged once into LDS with async b128 loads.
// ---------------------------------------------------------------------------
__global__ void attn_score_kernel(const float* __restrict__ enc,
                                  const float* __restrict__ WaBot,
                                  const float* __restrict__ hWa,
                                  const float* __restrict__ vvec,
                                  const int* __restrict__ mask,
                                  float* __restrict__ att) {
    __shared__ float lsA[16 * LDSPAD];                 // 16.6 KB
    __shared__ float s_att[16];

    int r0 = blockIdx.x << 4;            // flattened row base (s*B + b)
    int s0 = r0 >> 10;                   // /B
    int b0 = r0 & (Bn - 1);              // %B (tile never crosses s: 16 | 1024)

    if (threadIdx.x < 16) s_att[threadIdx.x] = 0.0f;

    // stage A tile: 16 rows x 64 chunks(16B) = 1024 chunks, 2 per thread
    for (int c = threadIdx.x; c < 16 * 64; c += 512) {
        int n  = c >> 6;
        int kc = (c & 63) << 2;
        async_cp16(&lsA[n * LDSPAD + kc], enc + (long)(r0 + n) * Hn + kc);
    }
    async_wait_all();
    __syncthreads();

    int n0   = (threadIdx.x >> 5) << 4;  // wave id * 16
    int lane = threadIdx.x & 31;
    int l    = lane & 15;
    int hl   = lane >> 4;

    v8f c = {};
    const float* arow = &lsA[l * LDSPAD];
    #pragma unroll 8
    for (int k0 = 0; k0 < Hn; k0 += 4) {
        int kk = k0 + 2 * hl;
        v2f a; a.x = arow[kk]; a.y = arow[kk + 1];
        const float* br = WaBot + (long)kk * Hn;
        v2f b; b.x = br[n0 + l]; b.y = br[Hn + n0 + l];
        c = __builtin_amdgcn_wmma_f32_16x16x4_f32(false, a, false, b, (short)0, c, false, false);
    }

    float vn = vvec[n0 + l];
    #pragma unroll
    for (int j = 0; j < 8; ++j) {
        int m = j + 8 * hl;              // output row within tile
        float e = tanhf(c[j] + hWa[(long)(b0 + m) * Hn + n0 + l]) * vn;
        // reduce over the 16 lanes of this half-wave (the N dimension)
        e += __shfl_xor(e, 8, 16);
        e += __shfl_xor(e, 4, 16);
        e += __shfl_xor(e, 2, 16);
        e += __shfl_xor(e, 1, 16);
        if (l == 0) atomicAdd(&s_att[m], e);
    }
    __syncthreads();

    if (threadIdx.x < 16) {
        int b = b0 + threadIdx.x;
        float val = s_att[threadIdx.x];
        if (mask[b * Sn + s0] == 0) val = -1000000.0f;
        att[(long)b * Sn + s0] = val;
    }
}

// ---------------------------------------------------------------------------
// Softmax over S per batch row (in place on att[B,S]).
// ---------------------------------------------------------------------------
__global__ void softmax_kernel(float* __restrict__ att) {
    __shared__ float red[256];
    int b = blockIdx.x, t = threadIdx.x;
    float x = (t < Sn) ? att[(long)b * Sn + t] : -3.0e38f;
    red[t] = x; __syncthreads();
    for (int s = 128; s > 0; s >>= 1) {
        if (t < s) red[t] = fmaxf(red[t], red[t + s]);
        __syncthreads();
    }
    float mx = red[0]; __syncthreads();
    float e = (t < Sn) ? __expf(x - mx) : 0.0f;
    red[t] = e; __syncthreads();
    for (int s = 128; s > 0; s >>= 1) {
        if (t < s) red[t] += red[t + s];
        __syncthreads();
    }
    float inv = 1.0f / red[0];
    if (t < Sn) att[(long)b * Sn + t] = e * inv;
}

// ---------------------------------------------------------------------------
// weighted[b,h] = sum_s a[b,s] * enc[s,b,h]   (coalesced over h)
// ---------------------------------------------------------------------------
__global__ void context_kernel(const float* __restrict__ a,
                               const float* __restrict__ enc,
                               float* __restrict__ weighted) {
    int idx = blockIdx.x * blockDim.x + threadIdx.x;   // B*H total
    int b = idx >> 8, h = idx & (Hn - 1);
    float acc = 0.0f;
    const float* arow = a + (long)b * Sn;
    #pragma unroll 4
    for (int s = 0; s < Sn; ++s)
        acc += arow[s] * enc[((long)s * Bn + b) * Hn + h];
    weighted[idx] = acc;
}

// ---------------------------------------------------------------------------
// rnn_in[b, 0:384] = [ emb[input_loc[b]], weighted[b] ]
// ---------------------------------------------------------------------------
__global__ void gather_rnn_in_kernel(const int* __restrict__ loc,
                                     const float* __restrict__ emb,
                                     const float* __restrict__ weighted,
                                     float* __restrict__ rnn_in) {
    int idx = blockIdx.x * blockDim.x + threadIdx.x;   // B*(E+H) total
    int b = idx / (En + Hn), k = idx % (En + Hn);
    float v = (k < En) ? emb[(long)loc[b] * En + k]
                       : weighted[(long)b * Hn + (k - En)];
    rnn_in[idx] = v;
}

// ---------------------------------------------------------------------------
// GRU combine: r,z,n gates -> h_new (written to ws and to tail of d_out)
// ---------------------------------------------------------------------------
__global__ void gru_combine_kernel(const float* __restrict__ gi,
                                   const float* __restrict__ gh,
                                   const float* __restrict__ h,
                                   float* __restrict__ hnew_ws,
                                   float* __restrict__ hnew_out) {
    int idx = blockIdx.x * blockDim.x + threadIdx.x;   // B*H total
    int b = idx >> 8, hh = idx & (Hn - 1);
    const float* gib = gi + (long)b * 3 * Hn;
    const float* ghb = gh + (long)b * 3 * Hn;
    float r = 1.0f / (1.0f + __expf(-(gib[hh] + ghb[hh])));
    float z = 1.0f / (1.0f + __expf(-(gib[Hn + hh] + ghb[Hn + hh])));
    float n = tanhf(gib[2 * Hn + hh] + r * ghb[2 * Hn + hh]);
    float out = (1.0f - z) * n + z * h[idx];
    hnew_ws[idx]  = out;
    hnew_out[idx] = out;
}

// ---------------------------------------------------------------------------
extern "C" void kernel_launch(void* const* d_in, const int* in_sizes, int n_in,
                              void* d_out, int out_size, void* d_ws, size_t ws_size,
                              hipStream_t stream) {
    const int*   input_loc = (const int*)  d_in[0];
    const float* hidden    = (const float*)d_in[1];   // (1,B,H) -> (B,H)
    const float* enc       = (const float*)d_in[2];   // (S,B,H)
    const int*   attn_mask = (const int*)  d_in[3];   // (B,S)
    const float* emb       = (const float*)d_in[4];   // (V,E)
    const float* Wa        = (const float*)d_in[5];   // (2H,H)
    const float* ba        = (const float*)d_in[6];   // (H,)
    const float* vvec      = (const float*)d_in[7];   // (H,)
    const float* W_ih      = (const float*)d_in[8];   // (3H, E+H)
    const float* W_hh      = (const float*)d_in[9];   // (3H, H)
    const float* b_ih      = (const float*)d_in[10];  // (3H,)
    const float* b_hh      = (const float*)d_in[11];  // (3H,)
    const float* W_out     = (const float*)d_in[12];  // (V,H)
    const float* b_out     = (const float*)d_in[13];  // (V,)

    float* pred = (float*)d_out;                          // (B,V)
    float* hnew_out = pred + (long)Bn * Vn;               // (B,H) tail of d_out

    // workspace carve-up (floats)
    float* ws      = (float*)d_ws;
    float* hWa     = ws;                      // B*H
    float* att     = hWa  + (long)Bn * Hn;    // B*S
    float* wgt     = att  + (long)Bn * Sn;    // B*H
    float* rnn_in  = wgt  + (long)Bn * Hn;    // B*(E+H)
    float* gi      = rnn_in + (long)Bn * (En + Hn); // B*3H
    float* gh      = gi   + (long)Bn * 3 * Hn;      // B*3H
    float* hnew_ws = gh   + (long)Bn * 3 * Hn;      // B*H

    // 1) hWa = h @ Wa_top + ba   (B-normal layout: Wa rows are K)
    {
        int M = Bn, N = Hn, K = Hn;
        int waves = (M / 16) * (N / 32);
        wmma_gemm_f32<false, false><<<(waves + 7) / 8, 256, 0, stream>>>(
            hidden, Wa, ba, hWa, M, N, K, Hn, Hn, (long)Hn, waves);
    }

    // 2) fused attention scores (enc @ Wa_bot, tanh, dot v, mask)
    attn_score_kernel<<<(Sn * Bn) / 16, 512, 0, stream>>>(
        enc, Wa + (long)Hn * Hn, hWa, vvec, attn_mask, att);

    // 3) softmax over S
    softmax_kernel<<<Bn, 256, 0, stream>>>(att);

    // 4) context vector
    context_kernel<<<(Bn * Hn) / 256, 256, 0, stream>>>(att, enc, wgt);

    // 5) assemble GRU input
    gather_rnn_in_kernel<<<(Bn * (En + Hn)) / 256, 256, 0, stream>>>(
        input_loc, emb, wgt, rnn_in);

    // 6) GRU gate GEMMs (B-transposed: weight rows are N)
    {
        int M = Bn, N = 3 * Hn, K = En + Hn;
        int waves = (M / 16) * (N / 32);
        wmma_gemm_f32<true, false><<<(waves + 7) / 8, 256, 0, stream>>>(
            rnn_in, W_ih, b_ih, gi, M, N, K, K, K, (long)N, waves);
    }
    {
        int M = Bn, N = 3 * Hn, K = Hn;
        int waves = (M / 16) * (N / 32);
        wmma_gemm_f32<true, false><<<(waves + 7) / 8, 256, 0, stream>>>(
            hidden, W_hh, b_hh, gh, M, N, K, K, K, (long)N, waves);
    }

    // 7) GRU combine -> h_new
    gru_combine_kernel<<<(Bn * Hn) / 256, 256, 0, stream>>>(
        gi, gh, hidden, hnew_ws, hnew_out);

    // 8) pred = h_new @ W_out.T + b_out  (async-LDS staged B, NT stores)
    {
        int blocks = (Bn / 128) * (Vn / 32);   // 8 * 3125 = 25000
        pred_gemm_kernel<<<blocks, 256, 0, stream>>>(hnew_ws, W_out, b_out, pred);
    }
}